// CaimanEmbedding_57750130262476
// MI455X (gfx1250) — compile-verified
//
#include <hip/hip_runtime.h>
#include <math.h>

// ---------------------------------------------------------------------------
// CDNA5 (gfx1250) implementation of the Caiman equivariant embedding.
// Node-side GEMMs use V_WMMA_F32_16X16X4_F32 (native fp32 matrix path,
// keeps reference precision). Edge scatter uses hardware fp32 global atomics
// into an L2-resident density array.
// ---------------------------------------------------------------------------

typedef float v2f __attribute__((ext_vector_type(2)));
typedef float v8f __attribute__((ext_vector_type(8)));

#define NELC 9
#define CCH  16
#define DDIM 128
#define NRBC 8

static __device__ inline v8f wmma_f32(v2f a, v2f b, v8f c) {
  // D = A(16x4) * B(4x16) + C(16x16), fp32, wave32
  return __builtin_amdgcn_wmma_f32_16x16x4_f32(false, a, false, b, (short)0, c,
                                               false, false);
}

static __device__ inline int lmap_of(int k) {  // LMAP = {0,1,1,1,2,2,2,2,2}
  return (k >= 4) ? 2 : ((k >= 1) ? 1 : 0);
}

// ------------------------------- zero fill ---------------------------------
__global__ void zero_f32_kernel(float* p, long n) {
  long i = (long)blockIdx.x * blockDim.x + threadIdx.x;
  long stride = (long)gridDim.x * blockDim.x;
  for (; i < n; i += stride) p[i] = 0.0f;
}

// ------------------------- CG tensor (quadrature) --------------------------
static __device__ inline double shc(int idx, double x, double y, double z) {
  const double S3 = 1.7320508075688772935;
  const double S5 = 2.2360679774997896964;
  const double S15 = 3.8729833462074168852;
  switch (idx) {
    case 0: return 1.0;
    case 1: return S3 * x;
    case 2: return S3 * y;
    case 3: return S3 * z;
    case 4: return S15 * x * y;
    case 5: return S15 * y * z;
    case 6: return 0.5 * S5 * (3.0 * z * z - 1.0);
    case 7: return S15 * x * z;
    default: return 0.5 * S15 * (x * x - y * y);
  }
}

__global__ void cg_init_kernel(float* cg) {
  int t = threadIdx.x;
  if (t >= 729) return;
  int i = t / 81, j = (t / 9) % 9, k = t % 9;
  // 12-point Gauss-Legendre nodes/weights on [-1,1]
  const double gx[6] = {0.1252334085114689, 0.3678314989981802,
                        0.5873179542866175, 0.7699026741943047,
                        0.9041172563704749, 0.9815606342467192};
  const double gw[6] = {0.2491470458134028, 0.2334925365383548,
                        0.2031674267230659, 0.1600783285433462,
                        0.1069393259953184, 0.0471753363865118};
  const double TWO_PI = 6.2831853071795864769;
  double acc = 0.0;
  for (int q = 0; q < 12; ++q) {
    double ct = (q < 6) ? gx[q] : -gx[q - 6];
    double w = (gw[q % 6] * 0.5) / 24.0;
    double st = sqrt(1.0 - ct * ct);
    for (int p = 0; p < 24; ++p) {
      double phi = (p + 0.5) * (TWO_PI / 24.0);
      double x = st * cos(phi), y = st * sin(phi), z = ct;
      acc += w * shc(i, x, y, z) * shc(j, x, y, z) * shc(k, x, y, z);
    }
  }
  cg[t] = (fabs(acc) < 1e-10) ? 0.0f : (float)acc;
}

// ----------------------- node embedding (WMMA GEMMs) -----------------------
// xi = species_table[species] @ W_se + b_se        (N,16)@(16,128)
__global__ void embed1_kernel(const float* __restrict__ st,
                              const int* __restrict__ species,
                              const float* __restrict__ W_se,
                              const float* __restrict__ b_se,
                              float* __restrict__ xi, int N) {
  int lane = threadIdx.x & 31;
  int half = lane >> 4, r = lane & 15;
  int nb = blockIdx.x * 16;
  bool full = (nb + 16 <= N);
  int rowc = min(nb + r, N - 1);
  int sp = species[rowc];
  float Arow[16];
#pragma unroll
  for (int i = 0; i < 16; ++i) Arow[i] = st[sp * 16 + i];
  for (int ct = 0; ct < 8; ++ct) {
    int col = ct * 16 + r;
    float bv = b_se[col];
    v8f acc = {bv, bv, bv, bv, bv, bv, bv, bv};
#pragma unroll
    for (int k0 = 0; k0 < 16; k0 += 4) {
      int ka = k0 + 2 * half;
      v2f a, b;
      a[0] = Arow[ka];
      a[1] = Arow[ka + 1];
      b[0] = W_se[ka * DDIM + col];
      b[1] = W_se[(ka + 1) * DDIM + col];
      acc = wmma_f32(a, b, acc);
    }
    float* o = xi + (long)(nb + 8 * half) * DDIM + col;
    if (full) {
#pragma unroll
      for (int v = 0; v < 8; ++v) o[(long)v * DDIM] = acc[v];
    } else {
#pragma unroll
      for (int v = 0; v < 8; ++v)
        if (nb + 8 * half + v < N) o[(long)v * DDIM] = acc[v];
    }
  }
}

// zsd = xi @ W_sl + b_sl       (N,128)@(128,32); cols 0..15 = Zs, 16..31 = Zd
__global__ void embed2_kernel(const float* __restrict__ xi,
                              const float* __restrict__ W_sl,
                              const float* __restrict__ b_sl,
                              float* __restrict__ zsd, int N) {
  int lane = threadIdx.x & 31;
  int half = lane >> 4, r = lane & 15;
  int nb = blockIdx.x * 16;
  bool full = (nb + 16 <= N);
  int rowc = min(nb + r, N - 1);
  const float* arow = xi + (long)rowc * DDIM + 2 * half;
  for (int ct = 0; ct < 2; ++ct) {
    int col = ct * 16 + r;
    float bv = b_sl[col];
    v8f acc = {bv, bv, bv, bv, bv, bv, bv, bv};
#pragma unroll
    for (int k0 = 0; k0 < DDIM; k0 += 4) {
      int ka = k0 + 2 * half;
      v2f a = *(const v2f*)(arow + k0);  // 8B-aligned, global_load_b64
      v2f b;
      b[0] = W_sl[ka * 32 + col];
      b[1] = W_sl[(ka + 1) * 32 + col];
      acc = wmma_f32(a, b, acc);
    }
    float* o = zsd + (long)(nb + 8 * half) * 32 + col;
    if (full) {
#pragma unroll
      for (int v = 0; v < 8; ++v) o[(long)v * 32] = acc[v];
    } else {
#pragma unroll
      for (int v = 0; v < 8; ++v)
        if (nb + 8 * half + v < N) o[(long)v * 32] = acc[v];
    }
  }
}

// ------------------------------ edge scatter -------------------------------
__global__ void edge_kernel(const float* __restrict__ dist,
                            const float* __restrict__ vec,
                            const float* __restrict__ sw,
                            const float* __restrict__ W_dij,
                            const float* __restrict__ b_dij,
                            const float* __restrict__ zsd,
                            const int* __restrict__ esrc,
                            const int* __restrict__ edst,
                            float* __restrict__ density, int E) {
  __shared__ float wd[NRBC * CCH];
  __shared__ float bd[CCH];
  for (int t = threadIdx.x; t < NRBC * CCH; t += blockDim.x) wd[t] = W_dij[t];
  for (int t = threadIdx.x; t < CCH; t += blockDim.x) bd[t] = b_dij[t];
  __syncthreads();
  int e = blockIdx.x * blockDim.x + threadIdx.x;
  if (e >= E) return;

  float d = dist[e];
  float invd = 1.0f / d;
  float s = sw[e];
  const float kpi = 3.14159265358979323846f / 5.0f;  // pi/CUTOFF
  const float amp = 0.632455532033675866f;           // sqrt(2/5)
  float rb[NRBC];
#pragma unroll
  for (int t = 0; t < NRBC; ++t)
    rb[t] = amp * __sinf((float)(t + 1) * kpi * d) * invd;

  float x = vec[3 * e] * invd, y = vec[3 * e + 1] * invd,
        z = vec[3 * e + 2] * invd;
  const float S3 = 1.7320508075688772f, S5 = 2.23606797749979f,
              S15 = 3.872983346207417f;
  float Y[NELC];
  Y[0] = 1.0f;
  Y[1] = S3 * x;
  Y[2] = S3 * y;
  Y[3] = S3 * z;
  Y[4] = S15 * x * y;
  Y[5] = S15 * y * z;
  Y[6] = 0.5f * S5 * (3.0f * z * z - 1.0f);
  Y[7] = S15 * x * z;
  Y[8] = 0.5f * S15 * (x * x - y * y);

  int src = esrc[e], dst = edst[e];
  float xij[CCH];
#pragma unroll
  for (int c = 0; c < CCH; ++c) {
    float acc = bd[c];
#pragma unroll
    for (int t = 0; t < NRBC; ++t) acc += rb[t] * wd[t * CCH + c];
    xij[c] = zsd[(long)src * 32 + c] * zsd[(long)dst * 32 + 16 + c] * acc * s;
  }
  float* drow = density + (long)src * (CCH * NELC);
#pragma unroll
  for (int c = 0; c < CCH; ++c) {
#pragma unroll
    for (int k = 0; k < NELC; ++k) {
      unsafeAtomicAdd(&drow[c * NELC + k], xij[c] * Y[k]);
    }
  }
}

// density[n,c,k] *= wsh[c, LMAP[k]]
__global__ void density_scale_kernel(float* __restrict__ density,
                                     const float* __restrict__ wsh, long total) {
  long i = (long)blockIdx.x * blockDim.x + threadIdx.x;
  long stride = (long)gridDim.x * blockDim.x;
  for (; i < total; i += stride) {
    int ck = (int)(i % (CCH * NELC));
    int c = ck / NELC, k = ck % NELC;
    density[i] *= wsh[c * 3 + lmap_of(k)];
  }
}

// ------------------------- channel mix (WMMA) ------------------------------
// out[n,o,m] = base[n,o,m]*acc_flag + 0.25 * sum_i w3[LMAP[m],o,i] * x[n,i,m]
__global__ void channel_mix_kernel(const float* __restrict__ w3,
                                   const float* __restrict__ x,
                                   const float* __restrict__ base,
                                   float* __restrict__ out, int N,
                                   int accumulate) {
  int lane = threadIdx.x & 31;
  int half = lane >> 4, r = lane & 15;
  int nb = blockIdx.x * 16;
  bool full = (nb + 16 <= N);
  int rowc = min(nb + r, N - 1);
  const float* xrow = x + (long)rowc * 144;
  for (int m = 0; m < NELC; ++m) {
    const float* w = w3 + lmap_of(m) * 256 + r * 16;
    v8f acc = {0, 0, 0, 0, 0, 0, 0, 0};
#pragma unroll
    for (int k0 = 0; k0 < CCH; k0 += 4) {
      int ka = k0 + 2 * half;
      v2f a, b;
      a[0] = xrow[ka * NELC + m];
      a[1] = xrow[(ka + 1) * NELC + m];
      b[0] = w[ka];        // B[k][n=o]=w[o][i]
      b[1] = w[ka + 1];
      acc = wmma_f32(a, b, acc);
    }
    long obase = (long)(nb + 8 * half) * 144 + r * NELC + m;
    if (full) {
#pragma unroll
      for (int v = 0; v < 8; ++v) {
        long o = obase + (long)v * 144;
        float val = 0.25f * acc[v];
        if (accumulate) val += base[o];
        out[o] = val;
      }
    } else {
#pragma unroll
      for (int v = 0; v < 8; ++v) {
        if (nb + 8 * half + v < N) {
          long o = obase + (long)v * 144;
          float val = 0.25f * acc[v];
          if (accumulate) val += base[o];
          out[o] = val;
        }
      }
    }
  }
}

// --------------------- filtered tensor product (sparse) --------------------
__global__ void ftp_kernel(const float* __restrict__ Vi,
                           const float* __restrict__ Hi,
                           const float* __restrict__ pw,   // (3,3,3,16)
                           const float* __restrict__ cg,   // (9,9,9)
                           float* __restrict__ Li, int N) {
  __shared__ float cgs[729];
  __shared__ float pws[27 * CCH];
  for (int t = threadIdx.x; t < 729; t += blockDim.x) cgs[t] = cg[t];
  for (int t = threadIdx.x; t < 27 * CCH; t += blockDim.x) pws[t] = pw[t];
  __syncthreads();
  long gid = (long)blockIdx.x * blockDim.x + threadIdx.x;
  if (gid >= (long)N * CCH) return;
  int n = (int)(gid >> 4);
  int c = (int)(gid & 15);
  float A[NELC], B[NELC], acc[NELC];
#pragma unroll
  for (int i = 0; i < NELC; ++i) {
    A[i] = Vi[(long)n * 144 + c * NELC + i];
    B[i] = Hi[(long)n * 144 + c * NELC + i];
    acc[i] = 0.0f;
  }
  for (int i = 0; i < NELC; ++i) {
    int li = lmap_of(i);
    for (int j = 0; j < NELC; ++j) {
      float ab = A[i] * B[j];
      int p0 = (li * 3 + lmap_of(j)) * 3;
#pragma unroll
      for (int k = 0; k < NELC; ++k) {
        float g = cgs[(i * 9 + j) * 9 + k];
        if (g != 0.0f) acc[k] += ab * g * pws[(p0 + lmap_of(k)) * CCH + c];
      }
    }
  }
#pragma unroll
  for (int k = 0; k < NELC; ++k) Li[(long)n * 144 + c * NELC + k] = acc[k];
}

// ----------------- fused latent MLP (WMMA through LDS) ---------------------
// h = silu([xi, Li[...,0]] @ W1 + b1); xi += h @ W2 + b2
__global__ void latent_kernel(const float* __restrict__ Li,
                              const float* __restrict__ W1,
                              const float* __restrict__ b1,
                              const float* __restrict__ W2,
                              const float* __restrict__ b2,
                              float* __restrict__ xi, int N) {
  __shared__ float As[16][146];  // even pad; 18*r injective mod 64
  __shared__ float Hs[16][130];  // even pad; 2*r injective mod 64
  int lane = threadIdx.x & 31;
  int half = lane >> 4, r = lane & 15;
  int nb = blockIdx.x * 16;
  bool full = (nb + 16 <= N);
  for (int t = threadIdx.x; t < 16 * 144; t += 32) {
    int row = t / 144, col = t % 144;
    int rc = min(nb + row, N - 1);
    As[row][col] = (col < DDIM) ? xi[(long)rc * DDIM + col]
                                : Li[(long)rc * 144 + (col - DDIM) * NELC];
  }
  __syncthreads();
  for (int ct = 0; ct < 8; ++ct) {
    int col = ct * 16 + r;
    float bv = b1[col];
    v8f acc = {bv, bv, bv, bv, bv, bv, bv, bv};
    const float* w1c = W1 + col;
#pragma unroll
    for (int k0 = 0; k0 < 144; k0 += 4) {
      int ka = k0 + 2 * half;
      v2f a = *(const v2f*)&As[r][ka];  // ds_load_b64
      v2f b;
      b[0] = w1c[(long)ka * DDIM];
      b[1] = w1c[(long)(ka + 1) * DDIM];
      acc = wmma_f32(a, b, acc);
    }
#pragma unroll
    for (int v = 0; v < 8; ++v) {
      float hx = acc[v];
      Hs[v + 8 * half][col] = hx / (1.0f + __expf(-hx));  // silu
    }
  }
  __syncthreads();
  for (int ct = 0; ct < 8; ++ct) {
    int col = ct * 16 + r;
    float bv = b2[col];
    v8f acc = {bv, bv, bv, bv, bv, bv, bv, bv};
    const float* w2c = W2 + col;
#pragma unroll
    for (int k0 = 0; k0 < DDIM; k0 += 4) {
      int ka = k0 + 2 * half;
      v2f a = *(const v2f*)&Hs[r][ka];  // ds_load_b64
      v2f b;
      b[0] = w2c[(long)ka * DDIM];
      b[1] = w2c[(long)(ka + 1) * DDIM];
      acc = wmma_f32(a, b, acc);
    }
    float* o = xi + (long)(nb + 8 * half) * DDIM + col;
    if (full) {
#pragma unroll
      for (int v = 0; v < 8; ++v) o[(long)v * DDIM] += acc[v];
    } else {
#pragma unroll
      for (int v = 0; v < 8; ++v)
        if (nb + 8 * half + v < N) o[(long)v * DDIM] += acc[v];
    }
  }
}

// ---------------------------------------------------------------------------
extern "C" void kernel_launch(void* const* d_in, const int* in_sizes, int n_in,
                              void* d_out, int out_size, void* d_ws,
                              size_t ws_size, hipStream_t stream) {
  const float* distances = (const float*)d_in[0];
  const float* vec = (const float*)d_in[1];
  const float* sw = (const float*)d_in[2];
  const float* species_table = (const float*)d_in[3];
  const float* W_se = (const float*)d_in[4];
  const float* b_se = (const float*)d_in[5];
  const float* W_dij = (const float*)d_in[6];
  const float* b_dij = (const float*)d_in[7];
  const float* W_sl = (const float*)d_in[8];
  const float* b_sl = (const float*)d_in[9];
  const float* wsh = (const float*)d_in[10];
  const float* cm0_w = (const float*)d_in[11];
  const float* dens_mix_w = (const float*)d_in[12];
  const float* tp_w = (const float*)d_in[13];
  const float* lW1 = (const float*)d_in[14];
  const float* lb1 = (const float*)d_in[15];
  const float* lW2 = (const float*)d_in[16];
  const float* lb2 = (const float*)d_in[17];
  const float* mix_w = (const float*)d_in[18];
  const int* species = (const int*)d_in[19];
  const int* esrc = (const int*)d_in[20];
  const int* edst = (const int*)d_in[21];

  const int E = in_sizes[0];
  const int N = in_sizes[19];
  const int NL = 3;

  float* ws = (float*)d_ws;
  long off = 0;
  float* density = ws + off; off += (long)N * 144;
  float* xi      = ws + off; off += (long)N * 128;
  float* zsd     = ws + off; off += (long)N * 32;
  float* Vi      = ws + off; off += (long)N * 144;
  float* Hi      = ws + off; off += (long)N * 144;
  float* Li      = ws + off; off += (long)N * 144;
  float* cg      = ws + off; off += 736;

  const int ntiles = (N + 15) / 16;
  const long dtot = (long)N * 144;
  const int zblocks = (int)((dtot + 255) / 256);

  zero_f32_kernel<<<zblocks, 256, 0, stream>>>(density, dtot);
  cg_init_kernel<<<1, 729, 0, stream>>>(cg);
  embed1_kernel<<<ntiles, 32, 0, stream>>>(species_table, species, W_se, b_se,
                                           xi, N);
  embed2_kernel<<<ntiles, 32, 0, stream>>>(xi, W_sl, b_sl, zsd, N);
  edge_kernel<<<(E + 255) / 256, 256, 0, stream>>>(
      distances, vec, sw, W_dij, b_dij, zsd, esrc, edst, density, E);
  density_scale_kernel<<<zblocks, 256, 0, stream>>>(density, wsh, dtot);
  channel_mix_kernel<<<ntiles, 32, 0, stream>>>(cm0_w, density, nullptr, Vi, N,
                                                0);
  const int fblocks = (int)(((long)N * 16 + 255) / 256);
  for (int l = 0; l < NL; ++l) {
    channel_mix_kernel<<<ntiles, 32, 0, stream>>>(dens_mix_w + (long)l * 768,
                                                  density, nullptr, Hi, N, 0);
    ftp_kernel<<<fblocks, 256, 0, stream>>>(Vi, Hi, tp_w + (long)l * 432, cg,
                                            Li, N);
    latent_kernel<<<ntiles, 32, 0, stream>>>(
        Li, lW1 + (long)l * 144 * 128, lb1 + (long)l * 128,
        lW2 + (long)l * 128 * 128, lb2 + (long)l * 128, xi, N);
    channel_mix_kernel<<<ntiles, 32, 0, stream>>>(mix_w + (long)l * 768, Li, Vi,
                                                  Vi, N, 1);
  }
  hipMemcpyAsync(d_out, xi, (size_t)N * 128 * sizeof(float),
                 hipMemcpyDeviceToDevice, stream);
  hipMemcpyAsync((float*)d_out + (long)N * 128, Vi,
                 (size_t)N * 144 * sizeof(float), hipMemcpyDeviceToDevice,
                 stream);
}